// Integrator_84310208020600
// MI455X (gfx1250) — compile-verified
//
#include <hip/hip_runtime.h>
#include <stdint.h>

// Geometry (from reference: XS=YS=ZS=256, F=8, N=1<<20)
#define VX   256
#define VV   16777216L          // 256^3 voxels
#define NUPD 1048576
#define FEAT 8

// Native clang vector (16B aligned) — required by __builtin_nontemporal_*
typedef __attribute__((ext_vector_type(4))) float v4f;
typedef __attribute__((ext_vector_type(4))) unsigned int v4u;

// d_out layout (tuple concatenated flat):
//   [0      ,   VV) new_v            (accumulates sum_u during scatter)
//   [VV     , 9*VV) new_f (VV*8)     (accumulates sum_uf during scatter)
//   [9*VV   ,10*VV) new_w            (accumulates sum_w during scatter)
//   [10*VV  ,11*VV) feature_weights  (accumulates sum_wf, overwritten at end)

__device__ __forceinline__ void atom_add_f32(float* p, float v) {
    // Hardware global_atomic_add_f32; result discarded -> no-return form (STOREcnt path)
    unsafeAtomicAdd(p, v);
}

// ---------------------------------------------------------------------------
// K1: zero the accumulator regions (all of d_out) + the 2*VV flag bytes in ws.
// ---------------------------------------------------------------------------
__global__ void k_init(v4f* __restrict__ out4, v4u* __restrict__ flags4,
                       long n_out4, long n_flag4) {
    long i = (long)blockIdx.x * blockDim.x + threadIdx.x;
    if (i < n_out4) {
        v4f z = {0.f, 0.f, 0.f, 0.f};
        out4[i] = z;
    }
    if (i < n_flag4) {
        v4u z = {0u, 0u, 0u, 0u};
        flags4[i] = z;
    }
}

// ---------------------------------------------------------------------------
// K2: scatter-add 1M updates into the accumulator regions of d_out.
// 11 hardware fp32 atomics + 2 idempotent byte stores per update; random
// across the volume but the hot footprint (~32MB of sum_uf) sits in the
// 192MB L2. Update streams are single-use -> non-temporal loads.
// ---------------------------------------------------------------------------
__global__ void k_scatter(const float* __restrict__ upd_vals,
                          const float* __restrict__ upd_feats,
                          const int*   __restrict__ upd_idx,
                          const int*   __restrict__ upd_fidx,
                          const float* __restrict__ upd_w,
                          float* __restrict__ out,
                          unsigned char* __restrict__ touched,
                          unsigned char* __restrict__ touchedf) {
    int n = blockIdx.x * blockDim.x + threadIdx.x;
    if (n >= NUPD) return;

    float w = __builtin_nontemporal_load(&upd_w[n]);

    // value / weight scatter
    int ax = upd_idx[3 * n + 0];
    int ay = upd_idx[3 * n + 1];
    int az = upd_idx[3 * n + 2];
    bool valid = ((unsigned)ax < VX) & ((unsigned)ay < VX) & ((unsigned)az < VX);
    if (valid) {
        long lin = ((long)ax * VX + ay) * VX + az;
        float v = __builtin_nontemporal_load(&upd_vals[n]);
        atom_add_f32(&out[lin], w * v);
        atom_add_f32(&out[9L * VV + lin], w);
        touched[lin] = 1;  // racy-but-idempotent
    }

    // feature scatter
    int bx = upd_fidx[3 * n + 0];
    int by = upd_fidx[3 * n + 1];
    int bz = upd_fidx[3 * n + 2];
    bool validf = ((unsigned)bx < VX) & ((unsigned)by < VX) & ((unsigned)bz < VX);
    if (validf) {
        long linf = ((long)bx * VX + by) * VX + bz;
        const v4f* f4 = (const v4f*)(upd_feats + (long)n * FEAT);  // 32B aligned
        v4f f0 = __builtin_nontemporal_load(&f4[0]);
        v4f f1 = __builtin_nontemporal_load(&f4[1]);
        float* dst = out + VV + linf * FEAT;
        atom_add_f32(dst + 0, w * f0.x);
        atom_add_f32(dst + 1, w * f0.y);
        atom_add_f32(dst + 2, w * f0.z);
        atom_add_f32(dst + 3, w * f0.w);
        atom_add_f32(dst + 4, w * f1.x);
        atom_add_f32(dst + 5, w * f1.y);
        atom_add_f32(dst + 6, w * f1.z);
        atom_add_f32(dst + 7, w * f1.w);
        atom_add_f32(&out[10L * VV + linf], w);
        touchedf[linf] = 1;
    }
}

// ---------------------------------------------------------------------------
// K3: per-voxel blend, in place over the accumulators.
// Both 32B/thread feature streams (features_volume f_old and the in-place
// sum_uf accumulator) are staged into LDS with the CDNA5 async copy engine
// (global_load_async_to_lds_b128, ASYNCcnt-tracked): 4 B128 DMA ops per
// thread, one s_wait_asynccnt covering all of them, fully overlapped with
// the scalar value/weight blend. Output is two non-temporal b128 stores
// per thread (536MB single-use stream).
// ---------------------------------------------------------------------------
__global__ void k_blend(const float* __restrict__ vals_vol,
                        const float* __restrict__ feats_vol,
                        const float* __restrict__ w_vol,
                        const float* __restrict__ fw_vol,
                        const int*   __restrict__ mw_p,
                        const unsigned char* __restrict__ touched,
                        const unsigned char* __restrict__ touchedf,
                        float* __restrict__ out) {
    __shared__ v4f tileF[256 * 2];  // f_old  : 8KB, 16B-aligned for ds_load_b128
    __shared__ v4f tileS[256 * 2];  // sum_uf : 8KB

    int  tid = threadIdx.x;
    long v   = (long)blockIdx.x * 256 + tid;  // grid sized exactly: EXEC all-1s

    float* fo = out + VV + v * FEAT;          // sum_uf in, new_f out (32B/thread)

    // Kick off 4 async global->LDS B128 copies. INST_OFFSET applies to both
    // the memory and LDS addresses, so each register pair + offset:16 covers
    // the thread's full 32B.
    unsigned long long gF = (unsigned long long)(uintptr_t)(feats_vol + v * FEAT);
    unsigned long long gS = (unsigned long long)(uintptr_t)fo;
    unsigned int ldsF = (unsigned int)(uintptr_t)(&tileF[0]) + (unsigned int)tid * 32u;
    unsigned int ldsS = (unsigned int)(uintptr_t)(&tileS[0]) + (unsigned int)tid * 32u;
    asm volatile("global_load_async_to_lds_b128 %0, %2, off\n\t"
                 "global_load_async_to_lds_b128 %0, %2, off offset:16\n\t"
                 "global_load_async_to_lds_b128 %1, %3, off\n\t"
                 "global_load_async_to_lds_b128 %1, %3, off offset:16"
                 :: "v"(ldsF), "v"(ldsS), "v"(gF), "v"(gS) : "memory");

    // Overlap: scalar (value/weight) blend while the DMA is in flight.
    float w_old  = __builtin_nontemporal_load(&w_vol[v]);
    float v_old  = __builtin_nontemporal_load(&vals_vol[v]);
    float fw_in  = __builtin_nontemporal_load(&fw_vol[v]);
    float sum_u  = out[v];
    float sum_w  = out[9L * VV + v];
    float sum_wf = out[10L * VV + v];
    bool  t  = touched[v]  != 0;
    bool  tf = touchedf[v] != 0;

    // max_weight scalar: tolerate int32 (500) or float32 bit patterns.
    int   mwi = mw_p[0];
    float mw  = (mwi >= 0 && mwi <= 1000000) ? (float)mwi : __int_as_float(mwi);

    float nv = t ? (w_old * v_old + sum_u) / (w_old + sum_w) : v_old;
    float nw = t ? fminf(fmaxf(w_old + sum_w, 0.0f), mw) : w_old;
    out[v]            = nv;
    out[9L * VV + v]  = nw;
    out[10L * VV + v] = fw_in;  // pass-through (sum_wf already read above)

    float invf = tf ? 1.0f / (w_old + sum_wf) : 0.0f;

    // Wait for all 4 async copies, then blend features from LDS (ds_load_b128).
    asm volatile("s_wait_asynccnt 0x0" ::: "memory");

    v4f f0 = tileF[tid * 2 + 0];
    v4f f1 = tileF[tid * 2 + 1];
    v4f s0 = tileS[tid * 2 + 0];
    v4f s1 = tileS[tid * 2 + 1];

    v4f o0, o1;
    o0.x = tf ? (w_old * f0.x + s0.x) * invf : f0.x;
    o0.y = tf ? (w_old * f0.y + s0.y) * invf : f0.y;
    o0.z = tf ? (w_old * f0.z + s0.z) * invf : f0.z;
    o0.w = tf ? (w_old * f0.w + s0.w) * invf : f0.w;
    o1.x = tf ? (w_old * f1.x + s1.x) * invf : f1.x;
    o1.y = tf ? (w_old * f1.y + s1.y) * invf : f1.y;
    o1.z = tf ? (w_old * f1.z + s1.z) * invf : f1.z;
    o1.w = tf ? (w_old * f1.w + s1.w) * invf : f1.w;

    v4f* fo4 = (v4f*)fo;  // 32B aligned (v*8 floats)
    __builtin_nontemporal_store(o0, &fo4[0]);
    __builtin_nontemporal_store(o1, &fo4[1]);
}

// ---------------------------------------------------------------------------
extern "C" void kernel_launch(void* const* d_in, const int* in_sizes, int n_in,
                              void* d_out, int out_size, void* d_ws, size_t ws_size,
                              hipStream_t stream) {
    const float* upd_vals  = (const float*)d_in[0];   // (1,1024,1024)        -> N
    const float* upd_feats = (const float*)d_in[1];   // (1,1024,1024,8)      -> N*8
    const int*   upd_idx   = (const int*)  d_in[2];   // (N,3)
    const int*   upd_fidx  = (const int*)  d_in[3];   // (N,3)
    /* d_in[4] filter_indices: unused by reference */
    const float* upd_w     = (const float*)d_in[5];   // (N,1)
    /* d_in[6], d_in[7]: *_empty, unused by reference */
    const float* vals_vol  = (const float*)d_in[8];   // (256,256,256)
    const float* feats_vol = (const float*)d_in[9];   // (256,256,256,8)
    const float* w_vol     = (const float*)d_in[10];  // (256,256,256)
    const float* fw_vol    = (const float*)d_in[11];  // (256,256,256)
    const int*   mw        = (const int*)  d_in[12];  // scalar max_weight

    float*         out      = (float*)d_out;
    unsigned char* touched  = (unsigned char*)d_ws;         // VV bytes
    unsigned char* touchedf = touched + VV;                 // VV bytes

    const long n_out4  = (11L * VV) / 4;   // v4f chunks covering all of d_out
    const long n_flag4 = (2L * VV) / 16;   // v4u chunks covering both flag arrays

    long init_blocks = (n_out4 + 255) / 256;
    k_init<<<(unsigned)init_blocks, 256, 0, stream>>>((v4f*)out, (v4u*)d_ws,
                                                      n_out4, n_flag4);

    k_scatter<<<NUPD / 256, 256, 0, stream>>>(upd_vals, upd_feats, upd_idx, upd_fidx,
                                              upd_w, out, touched, touchedf);

    k_blend<<<(unsigned)(VV / 256), 256, 0, stream>>>(vals_vol, feats_vol, w_vol, fw_vol,
                                                      mw, touched, touchedf, out);
}